// Memory_48722109006518
// MI455X (gfx1250) — compile-verified
//
#include <hip/hip_runtime.h>
#include <hip/hip_bf16.h>

typedef float v2f __attribute__((ext_vector_type(2)));
typedef float v8f __attribute__((ext_vector_type(8)));
typedef int   v4i_g __attribute__((vector_size(16)));   // matches async-LDS builtin params

#define PIX   16384   // B*H*W
#define CH    256     // KDIM == VDIM
#define NK    512     // NCLS*MSIZE
#define HWSZ  4096    // H*W
#define NCLS  8
#define MS    64

#define KC    32      // K chunk per LDS stage
#define LDP   36      // padded LDS row stride (floats): bank-conflict-free + 16B aligned rows

#if defined(__has_builtin)
#  if __has_builtin(__builtin_amdgcn_global_load_async_to_lds_b128)
#    define HAVE_ASYNC_LDS 1
#  endif
#endif
#ifndef HAVE_ASYNC_LDS
#  define HAVE_ASYNC_LDS 0
#endif
#if HAVE_ASYNC_LDS
#  define GLOBAL_AS __attribute__((address_space(1)))
#  define LDS_AS    __attribute__((address_space(3)))
#endif

// ---------------------------------------------------------------- utilities
__device__ __forceinline__ float wave_sum(float v) {
  for (int o = 16; o; o >>= 1) v += __shfl_xor(v, o, 32);
  return v;
}

__device__ __forceinline__ v8f wmma4(v2f a, v2f b, v8f c) {
  return __builtin_amdgcn_wmma_f32_16x16x4_f32(false, a, false, b, (short)0, c, false, false);
}

// ---------------------------------------------------------------- pack NCHW -> (P,256) rows + class id
__global__ void pack_kernel(const float* __restrict__ cont, const float* __restrict__ sty,
                            const float* __restrict__ mask, float* __restrict__ Q,
                            float* __restrict__ S, int* __restrict__ cls) {
  int p  = blockIdx.x * 256 + threadIdx.x;       // 64 blocks x 256
  int b  = p >> 12, hw = p & (HWSZ - 1);
  const float* cb = cont + (size_t)b * CH * HWSZ + hw;
  const float* sb = sty  + (size_t)b * CH * HWSZ + hw;
  float* q = Q + (size_t)p * CH;
  float* s = S + (size_t)p * CH;
  for (int d = 0; d < CH; ++d) {                  // coalesced reads across hw
    q[d] = cb[(size_t)d * HWSZ];
    s[d] = sb[(size_t)d * HWSZ];
  }
  const float* mb = mask + (size_t)b * NCLS * HWSZ + hw;
  int c = 0;
  for (int i = 1; i < NCLS; ++i)
    if (mb[(size_t)i * HWSZ] > 0.5f) c = i;       // one-hot
  cls[p] = c;
}

// ---------------------------------------------------------------- init accumulators
__global__ void init_kernel(const float* __restrict__ mk, const float* __restrict__ v1,
                            const float* __restrict__ v2, float* __restrict__ KU,
                            float* __restrict__ UV1, float* __restrict__ UV2,
                            float* __restrict__ lossblk) {
  int i = blockIdx.x * 256 + threadIdx.x;         // 512 blocks x 256 = NK*CH
  KU[i]  = 0.5f * mk[i];
  UV1[i] = 0.5f * v1[i];
  UV2[i] = 0.5f * v2[i];
  if (i < 64) lossblk[i] = 0.f;
}

// ---------------------------------------------------------------- fp32 WMMA GEMM:  C(MxN) = A(MxKd) * Bt(NxKd)^T
// LDS double-buffered via async global->LDS (ASYNCcnt), 8 waves, tile M=128 x N=64.
// mode 0: row-major C. mode 1: scatter to NCHW (B,256,64,64), Ntot==256.
__global__ __launch_bounds__(256) void gemm_nt_wmma(const float* __restrict__ A,
                                                    const float* __restrict__ Bt,
                                                    float* __restrict__ C,
                                                    int Kd, int Ntot, int mode) {
  __shared__ float As[2][128 * LDP];
  __shared__ float Bs[2][64 * LDP];

  const int tid  = threadIdx.x;
  const int lane = tid & 31;
  const int wv   = tid >> 5;                      // 0..7
  const int mt   = blockIdx.x * 128;
  const int nt   = blockIdx.y * 64;
  const int r    = lane & 15;
  const int kh   = (lane >> 4) * 2;               // K-pair select per 16x4 A layout

  const float* Abase = A  + (size_t)mt * Kd;
  const float* Bbase = Bt + (size_t)nt * Kd;

#if HAVE_ASYNC_LDS
  auto aissue = [&](int k0, int buf) {            // global -> LDS via ASYNCcnt path
#pragma unroll
    for (int j = 0; j < 4; ++j) {
      int i = tid + 256 * j;
      __builtin_amdgcn_global_load_async_to_lds_b128(
          (GLOBAL_AS v4i_g*)(Abase + (size_t)(i >> 3) * Kd + k0 + (i & 7) * 4),
          (LDS_AS v4i_g*)&As[buf][(i >> 3) * LDP + (i & 7) * 4], 0, 0);
    }
#pragma unroll
    for (int j = 0; j < 2; ++j) {
      int i = tid + 256 * j;
      __builtin_amdgcn_global_load_async_to_lds_b128(
          (GLOBAL_AS v4i_g*)(Bbase + (size_t)(i >> 3) * Kd + k0 + (i & 7) * 4),
          (LDS_AS v4i_g*)&Bs[buf][(i >> 3) * LDP + (i & 7) * 4], 0, 0);
    }
  };
#else
  float4 rA[4], rB[2];
  auto gload = [&](int k0) {                      // global -> regs (A: 1024 f4, B: 512 f4)
#pragma unroll
    for (int j = 0; j < 4; ++j) {
      int i = tid + 256 * j;
      rA[j] = *(const float4*)(Abase + (size_t)(i >> 3) * Kd + k0 + (i & 7) * 4);
    }
#pragma unroll
    for (int j = 0; j < 2; ++j) {
      int i = tid + 256 * j;
      rB[j] = *(const float4*)(Bbase + (size_t)(i >> 3) * Kd + k0 + (i & 7) * 4);
    }
  };
  auto sstore = [&](int buf) {                    // regs -> LDS (ds_store_b128)
#pragma unroll
    for (int j = 0; j < 4; ++j) {
      int i = tid + 256 * j;
      *(float4*)&As[buf][(i >> 3) * LDP + (i & 7) * 4] = rA[j];
    }
#pragma unroll
    for (int j = 0; j < 2; ++j) {
      int i = tid + 256 * j;
      *(float4*)&Bs[buf][(i >> 3) * LDP + (i & 7) * 4] = rB[j];
    }
  };
#endif

  v8f acc0 = {}; v8f acc1 = {}; v8f acc2 = {}; v8f acc3 = {};

  auto compute = [&](int buf) {                   // 8 k-steps x 4 WMMA from LDS
    const float* ar = &As[buf][(wv * 16 + r) * LDP + kh];
    const float* b0 = &Bs[buf][r * LDP + kh];
#pragma unroll
    for (int k = 0; k < KC; k += 4) {
      v2f a  = *(const v2f*)(ar + k);
      v2f q0 = *(const v2f*)(b0 + k);
      v2f q1 = *(const v2f*)(b0 + 16 * LDP + k);
      v2f q2 = *(const v2f*)(b0 + 32 * LDP + k);
      v2f q3 = *(const v2f*)(b0 + 48 * LDP + k);
      acc0 = wmma4(a, q0, acc0);
      acc1 = wmma4(a, q1, acc1);
      acc2 = wmma4(a, q2, acc2);
      acc3 = wmma4(a, q3, acc3);
    }
  };

  const int nch = Kd / KC;
#if HAVE_ASYNC_LDS
  aissue(0, 0);
  asm volatile("s_wait_asynccnt 0x0" ::: "memory");
  __syncthreads();
  for (int c = 0; c < nch; ++c) {
    int cur = c & 1;
    if (c + 1 < nch) aissue((c + 1) * KC, cur ^ 1);   // overlap DMA with WMMA chain
    compute(cur);
    asm volatile("s_wait_asynccnt 0x0" ::: "memory");
    __syncthreads();
  }
#else
  gload(0);
  sstore(0);
  __syncthreads();
  for (int c = 0; c < nch; ++c) {
    int cur = c & 1;
    if (c + 1 < nch) gload((c + 1) * KC);         // overlap global latency with WMMA chain
    compute(cur);
    if (c + 1 < nch) sstore(cur ^ 1);
    __syncthreads();
  }
#endif

  const int mhalf = 8 * (lane >> 4);              // C/D layout: VGPR rr -> M = rr + 8*(lane>=16)
#pragma unroll
  for (int j = 0; j < 4; ++j) {
    v8f acc = (j == 0) ? acc0 : (j == 1) ? acc1 : (j == 2) ? acc2 : acc3;
    int col = nt + j * 16 + r;
#pragma unroll
    for (int rr = 0; rr < 8; ++rr) {
      int row = mt + wv * 16 + rr + mhalf;
      float v = acc[rr];
      if (mode == 0) {
        C[(size_t)row * Ntot + col] = v;
      } else {                                    // unflat: (B,256,H,W)
        int b = row >> 12, hw = row & (HWSZ - 1);
        C[((size_t)(b * CH + col)) * HWSZ + hw] = v;
      }
    }
  }
}

// ---------------------------------------------------------------- masked column max over pixels of the column's class
__global__ void colmax_kernel(const float* __restrict__ SC, const int* __restrict__ cls,
                              float* __restrict__ colmax) {
  int j = blockIdx.x;                             // 512 columns
  int c = j >> 6;
  __shared__ float red[256];
  float m = -3e38f;
  for (int p = threadIdx.x; p < PIX; p += 256)
    if (cls[p] == c) m = fmaxf(m, SC[(size_t)p * NK + j]);
  red[threadIdx.x] = m;
  __syncthreads();
  for (int s = 128; s > 0; s >>= 1) {
    if (threadIdx.x < s) red[threadIdx.x] = fmaxf(red[threadIdx.x], red[threadIdx.x + s]);
    __syncthreads();
  }
  if (threadIdx.x == 0) colmax[j] = red[0];
}

// ---------------------------------------------------------------- per-pixel: class-block argmax, global top2, losses
// one wave per pixel. lossblk: [0:8]=n_a [8:16]=n_b [16:24]=mseK_a [24:32]=mseK_b
//                              [32:40]=mseV_a [40:48]=mseV_b [48]=trip_a [49]=trip_b
__global__ void analyze_kernel(const float* __restrict__ SC, const float* __restrict__ Q,
                               const float* __restrict__ S, const int* __restrict__ cls,
                               const float* __restrict__ keys, const float* __restrict__ vals,
                               int* __restrict__ ga, float* __restrict__ lossblk, int side) {
  int wid = threadIdx.x >> 5, lane = threadIdx.x & 31;
  int p = blockIdx.x * 8 + wid;
  int c = cls[p];
  int base = c * MS;
  const float* row = SC + (size_t)p * NK;

  float m1 = -3e38f, m2 = -3e38f; int i1 = 0x7fffffff, i2 = 0x7fffffff;
  float bm = -3e38f; int bi = base;
  for (int j = lane; j < NK; j += 32) {
    float v = row[j];
    if (v > m1) { m2 = m1; i2 = i1; m1 = v; i1 = j; }
    else if (v > m2) { m2 = v; i2 = j; }
    if (j >= base && j < base + MS && v > bm) { bm = v; bi = j; }
  }
  for (int o = 16; o; o >>= 1) {
    float om1 = __shfl_xor(m1, o, 32), om2 = __shfl_xor(m2, o, 32);
    int   oi1 = __shfl_xor(i1, o, 32), oi2 = __shfl_xor(i2, o, 32);
    float obm = __shfl_xor(bm, o, 32);
    int   obi = __shfl_xor(bi, o, 32);
    if (om1 > m1 || (om1 == m1 && oi1 < i1)) { m2 = m1; i2 = i1; m1 = om1; i1 = oi1; }
    else if (om1 > m2 || (om1 == m2 && oi1 < i2)) { m2 = om1; i2 = oi1; }
    if (om2 > m2 || (om2 == m2 && oi2 < i2)) { m2 = om2; i2 = oi2; }
    if (obm > bm || (obm == bm && obi < bi)) { bm = obm; bi = obi; }
  }
  int gl  = bi;
  int neg = (i1 == gl) ? i2 : i1;

  const float* q  = Q + (size_t)p * CH;
  const float* s  = S + (size_t)p * CH;
  const float* kp = keys + (size_t)gl  * CH;
  const float* kn = keys + (size_t)neg * CH;
  const float* vp = vals + (size_t)gl  * CH;
  float dp = 0.f, dn = 0.f, dv = 0.f;
  for (int d = lane; d < CH; d += 32) {
    float qd = q[d];
    float t  = qd - kp[d]; dp += t * t;
    t = qd - kn[d]; dn += t * t;
    t = s[d] - vp[d]; dv += t * t;
  }
  dp = wave_sum(dp); dn = wave_sum(dn); dv = wave_sum(dv);
  if (lane == 0) {
    ga[p] = gl;
    atomicAdd(&lossblk[side * 8 + c], 1.0f);
    atomicAdd(&lossblk[16 + side * 8 + c], dp);
    atomicAdd(&lossblk[32 + side * 8 + c], dv);
    atomicAdd(&lossblk[48 + side], fmaxf(sqrtf(dp) - sqrtf(dn) + 1.0f, 0.f));
  }
}

// ---------------------------------------------------------------- scatter update (one block per pixel)
__global__ void scatter_kernel(const float* __restrict__ Q, const float* __restrict__ S,
                               const int* __restrict__ ga, const float* __restrict__ colmax,
                               const float* __restrict__ SC, float* __restrict__ KU,
                               float* __restrict__ UV) {
  int p = blockIdx.x, d = threadIdx.x;
  int gl = ga[p];
  float w = __expf(SC[(size_t)p * NK + gl] - colmax[gl]);
  atomicAdd(&KU[(size_t)gl * CH + d], 0.25f * w * Q[(size_t)p * CH + d]);
  atomicAdd(&UV[(size_t)gl * CH + d], 0.50f * w * S[(size_t)p * CH + d]);
}

// ---------------------------------------------------------------- finalize: l2norm(KU)->UK + upd_* outputs + V^T builds
__global__ void finalize_kernel(const float* __restrict__ KU, const float* __restrict__ UV1,
                                const float* __restrict__ UV2, const float* __restrict__ r1,
                                const float* __restrict__ r2, float* __restrict__ UK,
                                float* __restrict__ V1T, float* __restrict__ V2T,
                                float* __restrict__ R1T, float* __restrict__ R2T,
                                float* __restrict__ out) {
  int j = blockIdx.x, d = threadIdx.x;            // 512 blocks x 256
  __shared__ float red[256];
  float ku = KU[(size_t)j * CH + d];
  red[d] = ku * ku;
  __syncthreads();
  for (int s = 128; s > 0; s >>= 1) {
    if (d < s) red[d] += red[d + s];
    __syncthreads();
  }
  float uk = ku / (sqrtf(red[0]) + 1e-12f);
  UK[(size_t)j * CH + d] = uk;
  out[(size_t)j * CH + d] = uk;                                // upd_keys
  float v1 = UV1[(size_t)j * CH + d], v2 = UV2[(size_t)j * CH + d];
  out[131072 + (size_t)j * CH + d] = v1;                       // upd_v1s
  out[262144 + (size_t)j * CH + d] = v2;                       // upd_v2s
  V1T[(size_t)d * NK + j] = v1;
  V2T[(size_t)d * NK + j] = v2;
  R1T[(size_t)d * NK + j] = r1[(size_t)j * CH + d];
  R2T[(size_t)d * NK + j] = r2[(size_t)j * CH + d];
}

// ---------------------------------------------------------------- in-place: RS row -> W row (class-block softmax, rest 0)
__global__ void build_w_kernel(float* __restrict__ SC, const int* __restrict__ cls) {
  int wid = threadIdx.x >> 5, lane = threadIdx.x & 31;
  int p = blockIdx.x * 8 + wid;
  int base = cls[p] * MS;
  float* row = SC + (size_t)p * NK;
  float v0 = row[base + lane], v1 = row[base + 32 + lane];
  float m = fmaxf(v0, v1);
  for (int o = 16; o; o >>= 1) m = fmaxf(m, __shfl_xor(m, o, 32));
  float e0 = __expf(v0 - m), e1 = __expf(v1 - m);
  float s = wave_sum(e0 + e1);
  float inv = 1.f / s;
  for (int j = lane; j < NK; j += 32) {
    float v = 0.f;
    if (j == base + lane)           v = e0 * inv;
    else if (j == base + 32 + lane) v = e1 * inv;
    row[j] = v;
  }
}

// ---------------------------------------------------------------- final scalar losses
__global__ void loss_final_kernel(const float* __restrict__ L, float* __restrict__ out) {
  if (threadIdx.x == 0 && blockIdx.x == 0) {
    float kl = 0.f, vl = 0.f;
    for (int c = 0; c < NCLS; ++c) {
      float na = fmaxf(L[c], 1.f), nb = fmaxf(L[8 + c], 1.f);
      kl += L[16 + c] / (na * 256.f) + L[24 + c] / (nb * 256.f);
      vl += L[32 + c] / (na * 256.f) + L[40 + c] / (nb * 256.f);
    }
    kl += (L[48] + L[49]) / 16384.f;
    out[33947648] = kl;
    out[33947649] = vl;
  }
}

// ================================================================ host
extern "C" void kernel_launch(void* const* d_in, const int* in_sizes, int n_in,
                              void* d_out, int out_size, void* d_ws, size_t ws_size,
                              hipStream_t stream) {
  const float* conts_a  = (const float*)d_in[0];
  const float* stys_a   = (const float*)d_in[1];
  const float* conts_b  = (const float*)d_in[2];
  const float* stys_b   = (const float*)d_in[3];
  const float* masks_a  = (const float*)d_in[4];
  const float* masks_b  = (const float*)d_in[5];
  const float* mem_keys = (const float*)d_in[6];   // (512,256) flat
  const float* mem_v1   = (const float*)d_in[7];
  const float* mem_v2   = (const float*)d_in[8];
  const float* rand1    = (const float*)d_in[9];
  const float* rand2    = (const float*)d_in[10];
  float* out = (float*)d_out;
  float* w = (float*)d_ws;

  // workspace layout (float offsets)
  float* Qa  = w;                        // P*256
  float* Qb  = Qa  + (size_t)PIX * CH;
  float* Sa  = Qb  + (size_t)PIX * CH;
  float* Sb  = Sa  + (size_t)PIX * CH;
  float* SCa = Sb  + (size_t)PIX * CH;   // P*512 (scores, then RS, then W)
  float* SCb = SCa + (size_t)PIX * NK;
  float* UK  = SCb + (size_t)PIX * NK;   // 512*256
  float* V1T = UK  + (size_t)NK * CH;    // 256*512
  float* V2T = V1T + (size_t)NK * CH;
  float* R1T = V2T + (size_t)NK * CH;
  float* R2T = R1T + (size_t)NK * CH;
  float* KU  = R2T + (size_t)NK * CH;
  float* UV1 = KU  + (size_t)NK * CH;
  float* UV2 = UV1 + (size_t)NK * CH;
  float* cmA = UV2 + (size_t)NK * CH;    // 512
  float* cmB = cmA + NK;
  float* LB  = cmB + NK;                 // 64
  int* clsA  = (int*)(LB + 64);
  int* clsB  = clsA + PIX;
  int* gaA   = clsB + PIX;
  int* gaB   = gaA  + PIX;

  // 1) pack + class ids + zero/seed accumulators
  pack_kernel<<<PIX / 256, 256, 0, stream>>>(conts_a, stys_a, masks_a, Qa, Sa, clsA);
  pack_kernel<<<PIX / 256, 256, 0, stream>>>(conts_b, stys_b, masks_b, Qb, Sb, clsB);
  init_kernel<<<NK, 256, 0, stream>>>(mem_keys, mem_v1, mem_v2, KU, UV1, UV2, LB);

  // 2) score GEMMs (all 8 classes + triplet keys at once): SC = Q @ keysT
  gemm_nt_wmma<<<dim3(PIX / 128, NK / 64), 256, 0, stream>>>(Qa, mem_keys, SCa, CH, NK, 0);
  gemm_nt_wmma<<<dim3(PIX / 128, NK / 64), 256, 0, stream>>>(Qb, mem_keys, SCb, CH, NK, 0);

  // 3) masked column maxima (softmax-over-pixels denominator cancellation)
  colmax_kernel<<<NK, 256, 0, stream>>>(SCa, clsA, cmA);
  colmax_kernel<<<NK, 256, 0, stream>>>(SCb, clsB, cmB);

  // 4) per-pixel argmax / top2 / mse + triplet partials
  analyze_kernel<<<PIX / 8, 256, 0, stream>>>(SCa, Qa, Sa, clsA, mem_keys, mem_v1, gaA, LB, 0);
  analyze_kernel<<<PIX / 8, 256, 0, stream>>>(SCb, Qb, Sb, clsB, mem_keys, mem_v2, gaB, LB, 1);

  // 5) scatter-update memories
  scatter_kernel<<<PIX, 256, 0, stream>>>(Qa, Sa, gaA, cmA, SCa, KU, UV1);
  scatter_kernel<<<PIX, 256, 0, stream>>>(Qb, Sb, gaB, cmB, SCb, KU, UV2);

  // 6) finalize UK (l2norm), emit upd_* outputs, build transposed value banks
  finalize_kernel<<<NK, 256, 0, stream>>>(KU, UV1, UV2, rand1, rand2, UK, V1T, V2T, R1T, R2T, out);
  loss_final_kernel<<<1, 64, 0, stream>>>(LB, out);

  // 7) rescore with updated keys: RS = Q @ UKT (reuse SC buffers)
  gemm_nt_wmma<<<dim3(PIX / 128, NK / 64), 256, 0, stream>>>(Qa, UK, SCa, CH, NK, 0);
  gemm_nt_wmma<<<dim3(PIX / 128, NK / 64), 256, 0, stream>>>(Qb, UK, SCb, CH, NK, 0);

  // 8) class-block softmax -> sparse mixing matrix W (in place)
  build_w_kernel<<<PIX / 8, 256, 0, stream>>>(SCa, clsA);
  build_w_kernel<<<PIX / 8, 256, 0, stream>>>(SCb, clsB);

  // 9) output GEMMs, stored directly in NCHW layout
  const size_t IMG = (size_t)4 * CH * HWSZ;       // 4194304
  float* sty_aa  = out + 393216;
  float* sty_ab  = sty_aa + IMG;
  float* sty_ba  = sty_ab + IMG;
  float* sty_bb  = sty_ba + IMG;
  float* rand_aa = sty_bb + IMG;
  float* rand_ab = rand_aa + IMG;
  float* rand_ba = rand_ab + IMG;
  float* rand_bb = rand_ba + IMG;

  gemm_nt_wmma<<<dim3(PIX / 128, CH / 64), 256, 0, stream>>>(SCa, V1T, sty_aa, NK, CH, 1);
  gemm_nt_wmma<<<dim3(PIX / 128, CH / 64), 256, 0, stream>>>(SCa, V2T, sty_ab, NK, CH, 1);
  gemm_nt_wmma<<<dim3(PIX / 128, CH / 64), 256, 0, stream>>>(SCa, R1T, rand_aa, NK, CH, 1);
  gemm_nt_wmma<<<dim3(PIX / 128, CH / 64), 256, 0, stream>>>(SCb, V1T, sty_ba, NK, CH, 1);
  gemm_nt_wmma<<<dim3(PIX / 128, CH / 64), 256, 0, stream>>>(SCb, V2T, sty_bb, NK, CH, 1);
  gemm_nt_wmma<<<dim3(PIX / 128, CH / 64), 256, 0, stream>>>(SCb, R2T, rand_ba, NK, CH, 1);
  // rand_ab == rand_aa and rand_bb == rand_ba in the reference
  (void)hipMemcpyAsync(rand_ab, rand_aa, IMG * sizeof(float), hipMemcpyDeviceToDevice, stream);
  (void)hipMemcpyAsync(rand_bb, rand_ba, IMG * sizeof(float), hipMemcpyDeviceToDevice, stream);
}